// GNN_EBM_Layer_node_51874615001420
// MI455X (gfx1250) — compile-verified
//
#include <hip/hip_runtime.h>

// ---------------------------------------------------------------------------
// GNN_EBM_Layer_node for MI455X (gfx1250, wave32)
//
// Phase 1 (scatter):  acc[b, edge0[m], c, :] += A[b,m] * x1[b, edge1[m], c, :]
// Phase 2 (WMMA GEMM): out[r,:] = (x1[r,:] + acc[r,:]) @ W^T + bias
//   rows r = flattened (b,t,c), R = 40000, K = N = 128, exact fp32 via
//   V_WMMA_F32_16X16X4_F32 (16x16 tile, K-step 4).
// ---------------------------------------------------------------------------

typedef float v2f __attribute__((ext_vector_type(2)));
typedef float v8f __attribute__((ext_vector_type(8)));

constexpr int kB    = 2;
constexpr int kT    = 10000;
constexpr int kM    = 100000;
constexpr int kD    = 128;
constexpr int kDOUT = 128;
constexpr int kLDSStride = 132;   // 128 + 4 pad: float4-aligned, conflict-free

__device__ __forceinline__ void atomicAddF32(float* p, float v) {
  unsafeAtomicAdd(p, v);          // lowers to global_atomic_add_f32
}

// ---------------------------------------------------------------------------
// Scatter: one wave32 per (b, c, m) row; lane l handles float4 chunk l.
// ---------------------------------------------------------------------------
__global__ __launch_bounds__(256) void scatter_kernel(
    const float* __restrict__ x1,   // (B, T, 2, D)
    const int*   __restrict__ edge, // (2, M): edge[m]=node_in, edge[M+m]=node_out
    const float* __restrict__ Am,   // (B, 1, M)
    float*       __restrict__ acc)  // (B, T, 2, D)
{
  const int wave = blockIdx.x * (blockDim.x >> 5) + (threadIdx.x >> 5);
  const int lane = threadIdx.x & 31;
  if (wave >= 4 * kM) return;

  const int bc = wave & 3;          // b,c vary fastest -> same edge in adjacent waves
  const int m  = wave >> 2;
  const int c  = bc & 1;
  const int b  = bc >> 1;

  const int nin  = edge[m];         // node_in  = edge[0][m]
  const int nout = edge[kM + m];    // node_out = edge[1][m]
  const float s  = Am[b * kM + m];

  const float4 v = *(const float4*)(x1 + ((size_t)((b * kT + nout) * 2 + c)) * kD + lane * 4);
  float* dst = acc + ((size_t)((b * kT + nin) * 2 + c)) * kD + lane * 4;

  atomicAddF32(dst + 0, v.x * s);
  atomicAddF32(dst + 1, v.y * s);
  atomicAddF32(dst + 2, v.z * s);
  atomicAddF32(dst + 3, v.w * s);
}

// ---------------------------------------------------------------------------
// GEMM: each wave computes a 16-row strip x all 128 output columns
// (8 accumulator tiles of 16x16), K swept in steps of 4 with f32 WMMA.
// ---------------------------------------------------------------------------
__global__ __launch_bounds__(256) void gemm_wmma_kernel(
    const float* __restrict__ x1,   // (R, D) flattened (b,t,c)
    const float* __restrict__ acc,  // (R, D)
    const float* __restrict__ W,    // (DOUT, D) row-major
    const float* __restrict__ bias, // (DOUT,)
    float*       __restrict__ out)  // (R, DOUT)
{
  extern __shared__ float lW[];     // DOUT x kLDSStride floats (67.6 KB)

  const int tid = threadIdx.x;

  // Stage W into LDS (padded rows): 4096 float4s across 256 threads.
  for (int i = tid; i < kDOUT * (kD / 4); i += 256) {
    const int row = i >> 5;         // / (D/4)
    const int c4  = (i & 31) * 4;
    const float4 wv = *(const float4*)(W + row * kD + c4);
    *(float4*)(&lW[row * kLDSStride + c4]) = wv;
  }
  __syncthreads();

  const int wid     = tid >> 5;
  const int lane    = tid & 31;
  const int strip   = blockIdx.x * 8 + wid;
  const int nstrips = (kB * kT * 2) / 16;       // 2500 (exact)
  if (strip >= nstrips) return;

  const int r0     = strip * 16;
  const int row    = r0 + (lane & 15);          // A-matrix row this lane feeds
  const int koff   = 2 * (lane >> 4);           // lane group selects K pair
  const int col15  = lane & 15;                 // B/C/D column within a tile

  const float* xrow = x1  + (size_t)row * kD;
  const float* arow = acc + (size_t)row * kD;

  v8f accv[8];
#pragma unroll
  for (int ot = 0; ot < 8; ++ot) {
    const float bv = bias[ot * 16 + col15];     // every C element in a lane shares N
#pragma unroll
    for (int i = 0; i < 8; ++i) accv[ot][i] = bv;
  }

  for (int k0 = 0; k0 < kD; k0 += 4) {
    const int k = k0 + koff;
    const float2 xv = *(const float2*)(xrow + k);
    const float2 av = *(const float2*)(arow + k);
    v2f a;
    a.x = xv.x + av.x;
    a.y = xv.y + av.y;

#pragma unroll
    for (int ot = 0; ot < 8; ++ot) {
      const float2 wv = *(const float2*)(&lW[(ot * 16 + col15) * kLDSStride + k]);
      v2f bf;
      bf.x = wv.x;
      bf.y = wv.y;
      // 8-arg form: (neg_a, A, neg_b, B, c_mod, C, reuse_a, reuse_b)
      accv[ot] = __builtin_amdgcn_wmma_f32_16x16x4_f32(
          false, a, false, bf, (short)0, accv[ot], false, false);
    }
  }

  // C/D layout: VGPR i -> row r0 + i + 8*(lane>>4), col = tile*16 + (lane&15)
  const int rbase = r0 + 8 * (lane >> 4);
#pragma unroll
  for (int ot = 0; ot < 8; ++ot) {
#pragma unroll
    for (int i = 0; i < 8; ++i) {
      out[(size_t)(rbase + i) * kDOUT + ot * 16 + col15] = accv[ot][i];
    }
  }
}

// ---------------------------------------------------------------------------
extern "C" void kernel_launch(void* const* d_in, const int* in_sizes, int n_in,
                              void* d_out, int out_size, void* d_ws, size_t ws_size,
                              hipStream_t stream) {
  (void)in_sizes; (void)n_in; (void)out_size; (void)ws_size;

  const float* x1   = (const float*)d_in[0];   // (B,T,2,D)
  // d_in[1] = x_2nd : unused by the reference
  const int*   edge = (const int*)  d_in[2];   // (2,M)
  const float* Am   = (const float*)d_in[3];   // (B,1,M)
  const float* W    = (const float*)d_in[4];   // (DOUT,D)
  const float* bias = (const float*)d_in[5];   // (DOUT,)
  float*       out  = (float*)d_out;           // (B,T,2,DOUT)
  float*       acc  = (float*)d_ws;            // (B,T,2,D) scratch accumulator

  const size_t accBytes = (size_t)kB * kT * 2 * kD * sizeof(float);
  hipMemsetAsync(acc, 0, accBytes, stream);

  // 400000 waves, 8 waves per 256-thread block -> 50000 blocks
  scatter_kernel<<<(4 * kM + 7) / 8, 256, 0, stream>>>(x1, edge, Am, acc);

  // 2500 strips, 8 waves per block -> 313 blocks; 67.6 KB dynamic LDS for W
  const size_t ldsBytes = (size_t)kDOUT * kLDSStride * sizeof(float);
  gemm_wmma_kernel<<<(2500 + 7) / 8, 256, ldsBytes, stream>>>(x1, acc, W, bias, out);
}